// MultiHeadSelfAttention_12884901888124
// MI455X (gfx1250) — compile-verified
//
#include <hip/hip_runtime.h>
#include <cstddef>
#include <cstdint>

#define BATCH    2
#define T_SEQ    2048
#define D_MODEL  768
#define N_HEADS  12
#define HEAD_DIM 64
#define D3       (3 * D_MODEL)

typedef __attribute__((ext_vector_type(2))) float v2f;
typedef __attribute__((ext_vector_type(8))) float v8f;

__device__ __forceinline__ v8f wmma_f32_4(v2f a, v2f b, v8f c) {
  // V_WMMA_F32_16X16X4_F32 : D(16x16,f32) = A(16x4,f32) x B(4x16,f32) + C
  return __builtin_amdgcn_wmma_f32_16x16x4_f32(false, a, false, b, (short)0, c,
                                               false, false);
}

// ---------------------------------------------------------------------------
// Async global->LDS copy (GLOBAL_LOAD_ASYNC_TO_LDS_B128, ASYNCcnt-tracked).
// Builtin prototype (from hipcc diagnostic): pointee is int vector_size(16).
// Falls back to a synchronous VGPR round-trip if the builtin is unavailable.
// ---------------------------------------------------------------------------
#if __has_builtin(__builtin_amdgcn_global_load_async_to_lds_b128) && \
    __has_builtin(__builtin_amdgcn_s_wait_asynccnt)
#define HAS_ASYNC_LDS 1
#else
#define HAS_ASYNC_LDS 0
#endif

typedef int v4i_vs __attribute__((vector_size(4 * sizeof(int))));
typedef __attribute__((address_space(1))) v4i_vs* gptr_b128;
typedef __attribute__((address_space(3))) v4i_vs* lptr_b128;

__device__ __forceinline__ void copy_b128_to_lds(const float* gsrc,
                                                 float* ldst) {
#if HAS_ASYNC_LDS
  __builtin_amdgcn_global_load_async_to_lds_b128((gptr_b128)gsrc,
                                                 (lptr_b128)ldst,
                                                 /*offset=*/0, /*cpol=*/0);
#else
  *reinterpret_cast<float4*>(ldst) = *reinterpret_cast<const float4*>(gsrc);
#endif
}

__device__ __forceinline__ void async_wait_all() {
#if HAS_ASYNC_LDS
  __builtin_amdgcn_s_wait_asynccnt(0);
#endif
}

// ---------------------------------------------------------------------------
// Tiled WMMA GEMM: C[M,N] = A[M,K] * B[K,N] (+ bias[N])
// Block tile 128x64, BK=16, 256 threads = 8 waves (4x2), wave tile 32x32.
// Double-buffered LDS, async DMA overlapped with WMMA on the other buffer.
// ---------------------------------------------------------------------------
template <bool WITH_BIAS>
__global__ __launch_bounds__(256) void gemm_wmma_kernel(
    const float* __restrict__ A, const float* __restrict__ Bm,
    const float* __restrict__ bias, float* __restrict__ C, int M, int N,
    int K) {
  const int tid = threadIdx.x;
  const int w = tid >> 5;
  const int lane = tid & 31;
  const int hh = lane >> 4;   // half-wave (K-group / M-group select)
  const int l15 = lane & 15;
  const int wr = w >> 1;      // wave row 0..3
  const int wc = w & 1;       // wave col 0..1
  const int bm0 = blockIdx.y * 128;
  const int bn0 = blockIdx.x * 64;

  // stride 20 floats = 80B rows: b128-aligned stores, conflict-free A reads
  __shared__ float As[2][128 * 20];
  __shared__ float Bs[2][16 * 64];

  v8f acc[2][2] = {};

  auto stage = [&](int kt, int buf) {
    // A tile 128x16 (512 x b128, 2 per thread)
#pragma unroll
    for (int it = 0; it < 2; ++it) {
      const int idx = tid + it * 256;
      const int r = idx >> 2;
      const int c4 = idx & 3;
      copy_b128_to_lds(A + (size_t)(bm0 + r) * K + kt * 16 + c4 * 4,
                       &As[buf][r * 20 + c4 * 4]);
    }
    // B tile 16x64 (256 x b128, 1 per thread)
    {
      const int r = tid >> 4;
      const int c4 = tid & 15;
      copy_b128_to_lds(Bm + (size_t)(kt * 16 + r) * N + bn0 + c4 * 4,
                       &Bs[buf][r * 64 + c4 * 4]);
    }
  };

  const int nk = K >> 4;
  stage(0, 0);
  async_wait_all();
  __syncthreads();

  for (int kt = 0; kt < nk; ++kt) {
    const int cur = kt & 1;
    if (kt + 1 < nk) stage(kt + 1, cur ^ 1);  // DMA next tile during compute

#pragma unroll
    for (int kk = 0; kk < 4; ++kk) {
      const int k0 = kk * 4 + hh * 2;
      v2f af[2], bf[2];
#pragma unroll
      for (int i = 0; i < 2; ++i) {
        const int r = wr * 32 + i * 16 + l15;
        af[i] = v2f{As[cur][r * 20 + k0], As[cur][r * 20 + k0 + 1]};
      }
#pragma unroll
      for (int j = 0; j < 2; ++j) {
        const int c = wc * 32 + j * 16 + l15;
        bf[j] = v2f{Bs[cur][k0 * 64 + c], Bs[cur][(k0 + 1) * 64 + c]};
      }
#pragma unroll
      for (int i = 0; i < 2; ++i)
#pragma unroll
        for (int j = 0; j < 2; ++j)
          acc[i][j] = wmma_f32_4(af[i], bf[j], acc[i][j]);
    }

    async_wait_all();
    __syncthreads();
  }

  // Store C from C/D layout: VGPR v -> M = hh*8+v, lane&15 -> N
#pragma unroll
  for (int i = 0; i < 2; ++i) {
#pragma unroll
    for (int j = 0; j < 2; ++j) {
      const int col = bn0 + wc * 32 + j * 16 + l15;
      const float bval = WITH_BIAS ? bias[col] : 0.0f;
#pragma unroll
      for (int v = 0; v < 8; ++v) {
        const int row = bm0 + wr * 32 + i * 16 + hh * 8 + v;
        C[(size_t)row * N + col] = acc[i][j][v] + bval;
      }
    }
  }
}

// ---------------------------------------------------------------------------
// Fused causal flash attention.
// Grid: (T/128, H, B). Block: 256 threads = 8 waves; wave w owns 16 Q rows.
// qkv layout: [B, T, 3*D]; output: [B, T, D] with head-major columns.
// K/V tiles double-buffered, async DMA overlapped with S/softmax/PV compute.
// ---------------------------------------------------------------------------
__global__ __launch_bounds__(256) void flash_attn_kernel(
    const float* __restrict__ qkv, float* __restrict__ out) {
  const int b = blockIdx.z;
  const int h = blockIdx.y;
  const int q0 = blockIdx.x * 128;
  const int tid = threadIdx.x;
  const int w = tid >> 5;
  const int lane = tid & 31;
  const int hh = lane >> 4;
  const int l15 = lane & 15;

  // stride 72 floats = 288B rows: b128-aligned, low-conflict B-fragment reads
  __shared__ float Ks[2][64 * 72];     // key tile   [64 keys][64 dims]
  __shared__ float Vs[2][64 * 72];     // value tile [64 keys][64 dims]
  __shared__ float Ps[8 * 16 * 68];    // per-wave P tile [16 rows][64 keys]
  float* Pw = &Ps[w * 16 * 68];

  const int qrow_base = q0 + w * 16;
  const float* qbase = qkv + (size_t)b * T_SEQ * D3 + h * HEAD_DIM;
  const float* kbase = qbase + D_MODEL;
  const float* vbase = qbase + 2 * D_MODEL;

  // Q as WMMA A-fragments, pre-scaled by 1/sqrt(64) = 0.125
  v2f qa[16];
  {
    const float* qp = qbase + (size_t)(qrow_base + l15) * D3;
#pragma unroll
    for (int kk = 0; kk < 16; ++kk) {
      const int k0 = kk * 4 + hh * 2;
      float2 t = *reinterpret_cast<const float2*>(qp + k0);
      qa[kk] = v2f{t.x * 0.125f, t.y * 0.125f};
    }
  }

  v8f o[4] = {};
  float mrow[8], lrow[8];
#pragma unroll
  for (int v = 0; v < 8; ++v) {
    mrow[v] = -3.0e38f;
    lrow[v] = 0.0f;
  }

  auto stageKV = [&](int kt, int buf) {
    // 64x64 K + 64x64 V, b128 granularity: 1024 transfers each, 4/thread
#pragma unroll
    for (int it = 0; it < 4; ++it) {
      const int idx = tid + it * 256;
      const int r = idx >> 4;
      const int c4 = (idx & 15) * 4;
      const size_t grow = (size_t)(kt * 64 + r) * D3;
      copy_b128_to_lds(kbase + grow + c4, &Ks[buf][r * 72 + c4]);
      copy_b128_to_lds(vbase + grow + c4, &Vs[buf][r * 72 + c4]);
    }
  };

  const int nkt = q0 / 64 + 2;  // key tiles for causal coverage of this block
  stageKV(0, 0);
  async_wait_all();
  __syncthreads();

  for (int kt = 0; kt < nkt; ++kt) {
    const int cur = kt & 1;
    const int kt0 = kt * 64;
    if (kt + 1 < nkt) stageKV(kt + 1, cur ^ 1);  // DMA next K/V during compute

    const bool active = (kt0 <= qrow_base + 15);  // wave-uniform
    if (active) {
      // --- S = (Q/8) * K^T : 4 N-tiles x 16 K-steps of K=4 ---
      v8f s[4] = {};
#pragma unroll
      for (int kk = 0; kk < 16; ++kk) {
        const int k0 = kk * 4 + hh * 2;
#pragma unroll
        for (int n = 0; n < 4; ++n) {
          const int c = n * 16 + l15;
          v2f bf = v2f{Ks[cur][c * 72 + k0], Ks[cur][c * 72 + k0 + 1]};
          s[n] = wmma_f32_4(qa[kk], bf, s[n]);
        }
      }
      // --- causal mask in C/D layout ---
#pragma unroll
      for (int n = 0; n < 4; ++n) {
        const int col = kt0 + n * 16 + l15;
#pragma unroll
        for (int v = 0; v < 8; ++v) {
          const int qr = qrow_base + hh * 8 + v;
          if (col > qr) s[n][v] = -3.0e38f;
        }
      }
      // --- online softmax: row max / rescale / row sum (16-lane reductions) ---
      float mnew[8], alpha[8];
#pragma unroll
      for (int v = 0; v < 8; ++v) {
        float mv = fmaxf(fmaxf(s[0][v], s[1][v]), fmaxf(s[2][v], s[3][v]));
        mv = fmaxf(mv, __shfl_xor(mv, 1, 32));
        mv = fmaxf(mv, __shfl_xor(mv, 2, 32));
        mv = fmaxf(mv, __shfl_xor(mv, 4, 32));
        mv = fmaxf(mv, __shfl_xor(mv, 8, 32));
        mnew[v] = fmaxf(mrow[v], mv);
        alpha[v] = __expf(mrow[v] - mnew[v]);
        mrow[v] = mnew[v];
      }
#pragma unroll
      for (int v = 0; v < 8; ++v) {
        float rs = 0.0f;
#pragma unroll
        for (int n = 0; n < 4; ++n) {
          const float p = __expf(s[n][v] - mnew[v]);
          s[n][v] = p;
          rs += p;
        }
        rs += __shfl_xor(rs, 1, 32);
        rs += __shfl_xor(rs, 2, 32);
        rs += __shfl_xor(rs, 4, 32);
        rs += __shfl_xor(rs, 8, 32);
        lrow[v] = lrow[v] * alpha[v] + rs;
      }
      // --- P: C/D layout -> LDS -> A layout ---
#pragma unroll
      for (int n = 0; n < 4; ++n)
#pragma unroll
        for (int v = 0; v < 8; ++v)
          Pw[(hh * 8 + v) * 68 + n * 16 + l15] = s[n][v];
      // --- rescale O, then O += P * V ---
#pragma unroll
      for (int n = 0; n < 4; ++n)
#pragma unroll
        for (int v = 0; v < 8; ++v) o[n][v] *= alpha[v];
#pragma unroll
      for (int kk = 0; kk < 16; ++kk) {
        const int k0 = kk * 4 + hh * 2;
        v2f pa = v2f{Pw[l15 * 68 + k0], Pw[l15 * 68 + k0 + 1]};
#pragma unroll
        for (int n = 0; n < 4; ++n) {
          const int c = n * 16 + l15;
          v2f vf = v2f{Vs[cur][k0 * 72 + c], Vs[cur][(k0 + 1) * 72 + c]};
          o[n] = wmma_f32_4(pa, vf, o[n]);
        }
      }
    }

    async_wait_all();
    __syncthreads();
  }

  // Normalize and store: out[b, qrow, h*64 + col]
  float inv[8];
#pragma unroll
  for (int v = 0; v < 8; ++v) inv[v] = 1.0f / lrow[v];
#pragma unroll
  for (int n = 0; n < 4; ++n) {
    const int col = h * HEAD_DIM + n * 16 + l15;
#pragma unroll
    for (int v = 0; v < 8; ++v) {
      const int row = qrow_base + hh * 8 + v;
      out[((size_t)b * T_SEQ + row) * D_MODEL + col] = o[n][v] * inv[v];
    }
  }
}

// ---------------------------------------------------------------------------
extern "C" void kernel_launch(void* const* d_in, const int* in_sizes, int n_in,
                              void* d_out, int out_size, void* d_ws,
                              size_t ws_size, hipStream_t stream) {
  const float* x = (const float*)d_in[0];
  // d_in[1] = attn_mask (causal tril) -- applied analytically in-kernel
  const float* Wqkv = (const float*)d_in[2];
  const float* Wo = (const float*)d_in[3];
  const float* bo = (const float*)d_in[4];
  float* out = (float*)d_out;

  float* qkv = (float*)d_ws;                          // [B*T, 3D]
  float* attnout = qkv + (size_t)BATCH * T_SEQ * D3;  // [B*T, D]

  const int M = BATCH * T_SEQ;  // 4096
  dim3 block(256);

  // 1) QKV projection: [4096, 2304] = x[4096, 768] @ Wqkv[768, 2304]
  dim3 g1(D3 / 64, M / 128);
  gemm_wmma_kernel<false><<<g1, block, 0, stream>>>(x, Wqkv, nullptr, qkv, M,
                                                    D3, D_MODEL);

  // 2) Fused causal flash attention
  dim3 g2(T_SEQ / 128, N_HEADS, BATCH);
  flash_attn_kernel<<<g2, block, 0, stream>>>(qkv, attnout);

  // 3) Output projection + bias: out = attnout @ Wo + bo
  dim3 g3(D_MODEL / 64, M / 128);
  gemm_wmma_kernel<true><<<g3, block, 0, stream>>>(attnout, Wo, bo, out, M,
                                                   D_MODEL, D_MODEL);
}